// VNSetAbstraction_60189671686866
// MI455X (gfx1250) — compile-verified
//
#include <hip/hip_runtime.h>

typedef __attribute__((ext_vector_type(2))) float v2f;
typedef __attribute__((ext_vector_type(8))) float v8f;

#define B_    4
#define N_    8192
#define CIN_  32
#define CO_   64
#define EDGE_ 65
#define EDGEP_ 68
#define M_    2048
#define K_    16
#define CNT_  131072.0f   /* B*M*k */

#if __has_builtin(__builtin_amdgcn_wmma_f32_16x16x4_f32)
#define HAVE_WMMA_X4 1
#endif

static __device__ inline v8f wmma_x4(v2f a, v2f b, v8f c) {
#ifdef HAVE_WMMA_X4
  // (neg_a, A, neg_b, B, c_mod, C, reuse_a, reuse_b)
  return __builtin_amdgcn_wmma_f32_16x16x4_f32(false, a, false, b, (short)0, c,
                                               false, false);
#else
  c[0] += a[0] * b[0] + a[1] * b[1];  // compile-safe placeholder (host pass)
  return c;
#endif
}

// ---------------------------------------------------------------------------
// 1) Farthest point sampling: one block per batch, points in registers.
// ---------------------------------------------------------------------------
__global__ __launch_bounds__(1024) void fps_kernel(
    const float* __restrict__ xyz, int* __restrict__ fpsIdx,
    float* __restrict__ outNewXyz) {
  int b = blockIdx.x;
  int tid = threadIdx.x;
  int lane = tid & 31, wid = tid >> 5;
  __shared__ int   ldsFps[M_];
  __shared__ float redV[32];
  __shared__ int   redI[32];
  __shared__ float ldsC[3];
  __shared__ int   ldsFar;

  const float* xb = xyz + (size_t)b * N_ * 3;
  float px[8], py[8], pz[8], dist[8];
#pragma unroll
  for (int j = 0; j < 8; j++) {
    int p = tid + j * 1024;
    px[j] = xb[p * 3 + 0];
    py[j] = xb[p * 3 + 1];
    pz[j] = xb[p * 3 + 2];
    dist[j] = 1e10f;
  }
  if (tid == 0) ldsFar = 0;
  __syncthreads();

  for (int it = 0; it < M_; it++) {
    int far = ldsFar;
    if (tid == 0) {
      ldsFps[it] = far;
      fpsIdx[b * M_ + it] = far;
      ldsC[0] = xb[far * 3 + 0];
      ldsC[1] = xb[far * 3 + 1];
      ldsC[2] = xb[far * 3 + 2];
    }
    __syncthreads();
    float cx = ldsC[0], cy = ldsC[1], cz = ldsC[2];
    float bv = -1.0f;
    int bi = 0;
#pragma unroll
    for (int j = 0; j < 8; j++) {
      float dx = px[j] - cx, dy = py[j] - cy, dz = pz[j] - cz;
      float d = dx * dx + dy * dy + dz * dz;
      float nd = fminf(dist[j], d);
      dist[j] = nd;
      if (nd > bv) { bv = nd; bi = tid + j * 1024; }
    }
    for (int off = 16; off > 0; off >>= 1) {
      float ov = __shfl_xor(bv, off, 32);
      int   oi = __shfl_xor(bi, off, 32);
      if (ov > bv || (ov == bv && oi < bi)) { bv = ov; bi = oi; }
    }
    if (lane == 0) { redV[wid] = bv; redI[wid] = bi; }
    __syncthreads();
    if (wid == 0) {
      bv = redV[lane];
      bi = redI[lane];
      for (int off = 16; off > 0; off >>= 1) {
        float ov = __shfl_xor(bv, off, 32);
        int   oi = __shfl_xor(bi, off, 32);
        if (ov > bv || (ov == bv && oi < bi)) { bv = ov; bi = oi; }
      }
      if (lane == 0) ldsFar = bi;
    }
    __syncthreads();
  }
  // write new_xyz (output tuple element 0)
  for (int m = tid; m < M_; m += 1024) {
    int idx = ldsFps[m];
    float* o = outNewXyz + ((size_t)b * M_ + m) * 3;
    o[0] = xb[idx * 3 + 0];
    o[1] = xb[idx * 3 + 1];
    o[2] = xb[idx * 3 + 2];
  }
}

// ---------------------------------------------------------------------------
// 2) KNN: one wave per query; per-lane sorted top-16, wave merge.
// ---------------------------------------------------------------------------
__global__ __launch_bounds__(256) void knn_kernel(
    const float* __restrict__ xyz, const float* __restrict__ newXyz,
    int* __restrict__ groupIdx) {
  int tid = threadIdx.x;
  int lane = tid & 31, wid = tid >> 5;
  int unit = blockIdx.x * 8 + wid;  // b*M + m
  int b = unit >> 11;
  const float* xb = xyz + (size_t)b * N_ * 3;
  float qx = newXyz[unit * 3 + 0];
  float qy = newXyz[unit * 3 + 1];
  float qz = newXyz[unit * 3 + 2];

  float bv[16];
  int   bi[16];
#pragma unroll
  for (int j = 0; j < 16; j++) { bv[j] = 3.0e38f; bi[j] = 0x7fffffff; }

  for (int p = lane; p < N_; p += 32) {
    float dx = xb[p * 3 + 0] - qx;
    float dy = xb[p * 3 + 1] - qy;
    float dz = xb[p * 3 + 2] - qz;
    float d = dx * dx + dy * dy + dz * dz;
    if (d < bv[15]) {
      int pos = 15;
      while (pos > 0 && bv[pos - 1] > d) {
        bv[pos] = bv[pos - 1];
        bi[pos] = bi[pos - 1];
        pos--;
      }
      bv[pos] = d;
      bi[pos] = p;
    }
  }
  int ptr = 0, mysel = 0;
  for (int r = 0; r < 16; r++) {
    float v  = (ptr < 16) ? bv[ptr] : 3.0e38f;
    int   pi = (ptr < 16) ? bi[ptr] : 0x7fffffff;
    float mv = v;
    int   mpi = pi;
    for (int off = 16; off > 0; off >>= 1) {
      float ov  = __shfl_xor(mv, off, 32);
      int   opi = __shfl_xor(mpi, off, 32);
      if (ov < mv || (ov == mv && opi < mpi)) { mv = ov; mpi = opi; }
    }
    if (v == mv && pi == mpi) ptr++;
    if (lane == r) mysel = mpi;
  }
  if (lane < 16) groupIdx[unit * 16 + lane] = mysel;
}

// ---------------------------------------------------------------------------
// Init: zero stats (128 f) + cov accumulators (36 f).
// ---------------------------------------------------------------------------
__global__ void init_kernel(float* __restrict__ stats, float* __restrict__ covb) {
  int t = threadIdx.x;
  if (t < 128) stats[t] = 0.0f;
  if (t < 36)  covb[t]  = 0.0f;
}

// ---------------------------------------------------------------------------
// Stage the 65x16x3 edge panel (padded to 68) for one (b,m) unit into LDS.
// Column n = neighbor index, row c = edge channel, plane v = vector dim.
// ---------------------------------------------------------------------------
static __device__ void stage_panel(const float* __restrict__ feat,
                                   const float* __restrict__ xyz,
                                   const float* __restrict__ newXyz,
                                   const int* __restrict__ fpsIdx,
                                   const int* __restrict__ groupIdx, int unit,
                                   int tid, int* ldsGi, float* ldsQ,
                                   float* ldsE) {
  int b = unit >> 11;
  if (tid < 16) ldsGi[tid] = groupIdx[unit * 16 + tid];
  if (tid == 16) ldsGi[16] = fpsIdx[unit];
  if (tid < 3) ldsQ[tid] = newXyz[unit * 3 + tid];
  __syncthreads();
  int fpsm = ldsGi[16];
  for (int i = tid; i < 3 * EDGEP_ * 16; i += 128) {
    int n = i & 15;
    int c = (i >> 4) % EDGEP_;
    int v = i / (EDGEP_ * 16);
    int gin = ldsGi[n];
    float val = 0.0f;
    if (c < 32) {
      const float* fp = feat + ((size_t)(b * 32 + c) * 3 + v) * N_;
      val = fp[gin] - fp[fpsm];
    } else if (c < 64) {
      const float* fp = feat + ((size_t)(b * 32 + (c - 32)) * 3 + v) * N_;
      val = fp[fpsm];
    } else if (c == 64) {
      val = xyz[((size_t)b * N_ + gin) * 3 + v] - ldsQ[v];
    }
    ldsE[(v * EDGEP_ + c) * 16 + n] = val;
  }
  __syncthreads();
}

// ---------------------------------------------------------------------------
// 3) BN stats pass: p = W_feat @ edge via WMMA f32 16x16x4; accumulate
//    sum / sumsq of (||p||+eps) per output channel. 8 m's per block.
// ---------------------------------------------------------------------------
__global__ __launch_bounds__(128) void stats_kernel(
    const float* __restrict__ feat, const float* __restrict__ xyz,
    const float* __restrict__ newXyz, const int* __restrict__ fpsIdx,
    const int* __restrict__ groupIdx, const float* __restrict__ Wf,
    float* __restrict__ stats) {
  __shared__ int   ldsGi[17];
  __shared__ float ldsQ[3];
  __shared__ float ldsE[3 * EDGEP_ * 16];
  int tid = threadIdx.x;
  int lane = tid & 31, wid = tid >> 5;
  int row = lane & 15, lh = lane >> 4;

  v2f aF[17];
#pragma unroll
  for (int s = 0; s < 17; s++) {
    int k0 = 4 * s + 2 * lh;
    int rw = wid * 16 + row;
    v2f a; a.x = 0.0f; a.y = 0.0f;
    if (k0 < EDGE_)     a.x = Wf[rw * EDGE_ + k0];
    if (k0 + 1 < EDGE_) a.y = Wf[rw * EDGE_ + k0 + 1];
    aF[s] = a;
  }
  float ns[8], nss[8];
#pragma unroll
  for (int j = 0; j < 8; j++) { ns[j] = 0.0f; nss[j] = 0.0f; }

  const v8f vzero = {0.f, 0.f, 0.f, 0.f, 0.f, 0.f, 0.f, 0.f};
  for (int mi = 0; mi < 8; mi++) {
    int unit = blockIdx.x * 8 + mi;
    stage_panel(feat, xyz, newXyz, fpsIdx, groupIdx, unit, tid, ldsGi, ldsQ, ldsE);
    v8f accP[3];
    accP[0] = vzero; accP[1] = vzero; accP[2] = vzero;
#pragma unroll
    for (int s = 0; s < 17; s++) {
      int k0 = 4 * s + 2 * lh;
#pragma unroll
      for (int v = 0; v < 3; v++) {
        v2f bf;
        bf.x = ldsE[(v * EDGEP_ + k0) * 16 + row];
        bf.y = ldsE[(v * EDGEP_ + k0 + 1) * 16 + row];
        accP[v] = wmma_x4(aF[s], bf, accP[v]);
      }
    }
#pragma unroll
    for (int j = 0; j < 8; j++) {
      float p0 = accP[0][j], p1 = accP[1][j], p2 = accP[2][j];
      float nrm = sqrtf(p0 * p0 + p1 * p1 + p2 * p2) + 1e-6f;
      ns[j] += nrm;
      nss[j] += nrm * nrm;
    }
  }
#pragma unroll
  for (int j = 0; j < 8; j++) {
    for (int off = 8; off > 0; off >>= 1) {
      ns[j]  += __shfl_xor(ns[j], off, 32);
      nss[j] += __shfl_xor(nss[j], off, 32);
    }
  }
  if (row == 0) {
#pragma unroll
    for (int j = 0; j < 8; j++) {
      int chan = wid * 16 + j + 8 * lh;
      atomicAdd(&stats[chan], ns[j]);
      atomicAdd(&stats[64 + chan], nss[j]);
    }
  }
}

// ---------------------------------------------------------------------------
// 4) Main pass: p & d via WMMA, VN-BN + VN-LeakyReLU + mean-pool over k.
//    One block per (b,m) unit; wave w owns channel tile w.
// ---------------------------------------------------------------------------
__global__ __launch_bounds__(128) void main_kernel(
    const float* __restrict__ feat, const float* __restrict__ xyz,
    const float* __restrict__ newXyz, const int* __restrict__ fpsIdx,
    const int* __restrict__ groupIdx, const float* __restrict__ Wf,
    const float* __restrict__ Wd, const float* __restrict__ stats,
    const float* __restrict__ bnw, const float* __restrict__ bnb,
    float* __restrict__ nf) {
  __shared__ int   ldsGi[17];
  __shared__ float ldsQ[3];
  __shared__ float ldsE[3 * EDGEP_ * 16];
  int tid = threadIdx.x;
  int lane = tid & 31, wid = tid >> 5;
  int row = lane & 15, lh = lane >> 4;
  int unit = blockIdx.x;
  int b = unit >> 11, m = unit & (M_ - 1);

  v2f aF[17], aD[17];
#pragma unroll
  for (int s = 0; s < 17; s++) {
    int k0 = 4 * s + 2 * lh;
    int rw = wid * 16 + row;
    v2f af; af.x = 0.0f; af.y = 0.0f;
    v2f ad; ad.x = 0.0f; ad.y = 0.0f;
    if (k0 < EDGE_)     { af.x = Wf[rw * EDGE_ + k0];     ad.x = Wd[rw * EDGE_ + k0]; }
    if (k0 + 1 < EDGE_) { af.y = Wf[rw * EDGE_ + k0 + 1]; ad.y = Wd[rw * EDGE_ + k0 + 1]; }
    aF[s] = af; aD[s] = ad;
  }
  stage_panel(feat, xyz, newXyz, fpsIdx, groupIdx, unit, tid, ldsGi, ldsQ, ldsE);

  const v8f vzero = {0.f, 0.f, 0.f, 0.f, 0.f, 0.f, 0.f, 0.f};
  v8f accP[3], accD[3];
#pragma unroll
  for (int v = 0; v < 3; v++) { accP[v] = vzero; accD[v] = vzero; }

#pragma unroll
  for (int s = 0; s < 17; s++) {
    int k0 = 4 * s + 2 * lh;
#pragma unroll
    for (int v = 0; v < 3; v++) {
      v2f bf;
      bf.x = ldsE[(v * EDGEP_ + k0) * 16 + row];
      bf.y = ldsE[(v * EDGEP_ + k0 + 1) * 16 + row];
      accP[v] = wmma_x4(aF[s], bf, accP[v]);
      accD[v] = wmma_x4(aD[s], bf, accD[v]);
    }
  }

#pragma unroll
  for (int j = 0; j < 8; j++) {
    int chan = wid * 16 + j + 8 * lh;
    float s  = stats[chan];
    float ss = stats[64 + chan];
    float mu = s * (1.0f / CNT_);
    float var = ss * (1.0f / CNT_) - mu * mu;
    float rstd = rsqrtf(var + 1e-5f);
    float bw = bnw[chan], bb = bnb[chan];

    float p0 = accP[0][j], p1 = accP[1][j], p2 = accP[2][j];
    float d0 = accD[0][j], d1 = accD[1][j], d2 = accD[2][j];
    float nrm = sqrtf(p0 * p0 + p1 * p1 + p2 * p2) + 1e-6f;
    float nbn = (nrm - mu) * rstd * bw + bb;
    float sc = nbn / nrm;
    p0 *= sc; p1 *= sc; p2 *= sc;
    float dotv = p0 * d0 + p1 * d1 + p2 * d2;
    float dsq = d0 * d0 + d1 * d1 + d2 * d2;
    float coef = (dotv >= 0.0f) ? 0.0f : (0.9f * dotv / (dsq + 1e-6f));
    float h0 = p0 - coef * d0;
    float h1 = p1 - coef * d1;
    float h2 = p2 - coef * d2;
    for (int off = 8; off > 0; off >>= 1) {  // mean over 16 neighbors
      h0 += __shfl_xor(h0, off, 32);
      h1 += __shfl_xor(h1, off, 32);
      h2 += __shfl_xor(h2, off, 32);
    }
    if (row == 0) {
      float* nb = nf + ((size_t)(b * CO_ + chan) * 3) * M_ + m;
      nb[0]        = h0 * (1.0f / 16.0f);
      nb[M_]       = h1 * (1.0f / 16.0f);
      nb[2 * M_]   = h2 * (1.0f / 16.0f);
    }
  }
}

// ---------------------------------------------------------------------------
// 5) ZCA: per-batch 3x3 covariance reduction, Jacobi eigensolve, whitening.
// ---------------------------------------------------------------------------
__global__ __launch_bounds__(256) void cov_kernel(const float* __restrict__ nf,
                                                  float* __restrict__ covb) {
  int b = blockIdx.x >> 6, o = blockIdx.x & 63;
  const float* p0 = nf + ((size_t)(b * CO_ + o) * 3 + 0) * M_;
  const float* p1 = p0 + M_;
  const float* p2 = p1 + M_;
  float acc[9];
#pragma unroll
  for (int j = 0; j < 9; j++) acc[j] = 0.0f;
  for (int m = threadIdx.x; m < M_; m += 256) {
    float x0 = p0[m], x1 = p1[m], x2 = p2[m];
    acc[0] += x0; acc[1] += x1; acc[2] += x2;
    acc[3] += x0 * x0; acc[4] += x0 * x1; acc[5] += x0 * x2;
    acc[6] += x1 * x1; acc[7] += x1 * x2; acc[8] += x2 * x2;
  }
#pragma unroll
  for (int j = 0; j < 9; j++)
    for (int off = 16; off > 0; off >>= 1) acc[j] += __shfl_xor(acc[j], off, 32);
  if ((threadIdx.x & 31) == 0) {
#pragma unroll
    for (int j = 0; j < 9; j++) atomicAdd(&covb[b * 9 + j], acc[j]);
  }
}

__global__ void zca_kernel(const float* __restrict__ covb,
                           float* __restrict__ wzb) {
  int t = threadIdx.x;
  if (t >= B_) return;
  const float* c = covb + t * 9;
  float Mt = (float)(CO_ * M_);
  float mean0 = c[0] / Mt, mean1 = c[1] / Mt, mean2 = c[2] / Mt;
  float inv = 1.0f / (Mt + 1e-6f);
  float A[3][3];
  A[0][0] = (c[3] - Mt * mean0 * mean0) * inv + 1e-5f;
  A[0][1] = A[1][0] = (c[4] - Mt * mean0 * mean1) * inv;
  A[0][2] = A[2][0] = (c[5] - Mt * mean0 * mean2) * inv;
  A[1][1] = (c[6] - Mt * mean1 * mean1) * inv + 1e-5f;
  A[1][2] = A[2][1] = (c[7] - Mt * mean1 * mean2) * inv;
  A[2][2] = (c[8] - Mt * mean2 * mean2) * inv + 1e-5f;
  float V[3][3] = {{1, 0, 0}, {0, 1, 0}, {0, 0, 1}};
  for (int sweep = 0; sweep < 10; sweep++) {
    for (int pq = 0; pq < 3; pq++) {
      int p = (pq == 2) ? 1 : 0;
      int q = (pq == 0) ? 1 : 2;
      float apq = A[p][q];
      if (fabsf(apq) < 1e-30f) continue;
      float theta = (A[q][q] - A[p][p]) / (2.0f * apq);
      float tt = ((theta >= 0.0f) ? 1.0f : -1.0f) /
                 (fabsf(theta) + sqrtf(theta * theta + 1.0f));
      float cc = 1.0f / sqrtf(tt * tt + 1.0f);
      float sn = tt * cc;
      for (int k = 0; k < 3; k++) {
        float akp = A[k][p], akq = A[k][q];
        A[k][p] = cc * akp - sn * akq;
        A[k][q] = sn * akp + cc * akq;
      }
      for (int k = 0; k < 3; k++) {
        float apk = A[p][k], aqk = A[q][k];
        A[p][k] = cc * apk - sn * aqk;
        A[q][k] = sn * apk + cc * aqk;
      }
      for (int k = 0; k < 3; k++) {
        float vkp = V[k][p], vkq = V[k][q];
        V[k][p] = cc * vkp - sn * vkq;
        V[k][q] = sn * vkp + cc * vkq;
      }
    }
  }
  float iw0 = rsqrtf(fmaxf(A[0][0], 1e-5f));
  float iw1 = rsqrtf(fmaxf(A[1][1], 1e-5f));
  float iw2 = rsqrtf(fmaxf(A[2][2], 1e-5f));
  float* wz = wzb + t * 12;
  for (int i = 0; i < 3; i++)
    for (int j = 0; j < 3; j++)
      wz[i * 3 + j] = V[i][0] * V[j][0] * iw0 + V[i][1] * V[j][1] * iw1 +
                      V[i][2] * V[j][2] * iw2;
  wz[9] = mean0; wz[10] = mean1; wz[11] = mean2;
}

__global__ __launch_bounds__(256) void out_kernel(
    const float* __restrict__ nf, const float* __restrict__ wzb,
    const float* __restrict__ gamma, float* __restrict__ outFeat) {
  int idx = blockIdx.x * 256 + threadIdx.x;  // < B*CO*M = 2^19
  int m = idx & (M_ - 1);
  int o = (idx >> 11) & 63;
  int b = idx >> 17;
  const float* wz = wzb + b * 12;
  const float* base = nf + ((size_t)(b * CO_ + o) * 3) * M_;
  float x0 = base[m]          - wz[9];
  float x1 = base[M_ + m]     - wz[10];
  float x2 = base[2 * M_ + m] - wz[11];
  float g = gamma[o];
  float* ob = outFeat + ((size_t)(b * CO_ + o) * 3) * M_;
  ob[m]          = g * (wz[0] * x0 + wz[1] * x1 + wz[2] * x2);
  ob[M_ + m]     = g * (wz[3] * x0 + wz[4] * x1 + wz[5] * x2);
  ob[2 * M_ + m] = g * (wz[6] * x0 + wz[7] * x1 + wz[8] * x2);
}

// ---------------------------------------------------------------------------
extern "C" void kernel_launch(void* const* d_in, const int* in_sizes, int n_in,
                              void* d_out, int out_size, void* d_ws,
                              size_t ws_size, hipStream_t stream) {
  (void)in_sizes; (void)n_in; (void)out_size; (void)ws_size;
  const float* xyz   = (const float*)d_in[0];
  const float* feat  = (const float*)d_in[1];
  const float* Wf    = (const float*)d_in[2];
  const float* Wd    = (const float*)d_in[3];
  const float* bnw   = (const float*)d_in[4];
  const float* bnb   = (const float*)d_in[5];
  const float* gamma = (const float*)d_in[6];

  float* out        = (float*)d_out;
  float* outNewXyz  = out;              // (B, M, 3)
  float* outFeat    = out + B_ * M_ * 3;  // (B, CO, 3, M)

  int*   fpsIdx   = (int*)d_ws;                 // B*M
  int*   groupIdx = fpsIdx + B_ * M_;           // B*M*K
  float* stats    = (float*)(groupIdx + B_ * M_ * K_);  // 128
  float* covb     = stats + 128;                // 36
  float* wzb      = covb + 36;                  // 48
  float* nf       = wzb + 48;                   // B*CO*3*M

  fps_kernel<<<B_, 1024, 0, stream>>>(xyz, fpsIdx, outNewXyz);
  knn_kernel<<<B_ * M_ / 8, 256, 0, stream>>>(xyz, outNewXyz, groupIdx);
  init_kernel<<<1, 256, 0, stream>>>(stats, covb);
  stats_kernel<<<B_ * M_ / 8, 128, 0, stream>>>(feat, xyz, outNewXyz, fpsIdx,
                                                groupIdx, Wf, stats);
  main_kernel<<<B_ * M_, 128, 0, stream>>>(feat, xyz, outNewXyz, fpsIdx,
                                           groupIdx, Wf, Wd, stats, bnw, bnb,
                                           nf);
  cov_kernel<<<B_ * CO_, 256, 0, stream>>>(nf, covb);
  zca_kernel<<<1, 64, 0, stream>>>(covb, wzb);
  out_kernel<<<(B_ * CO_ * M_) / 256, 256, 0, stream>>>(nf, wzb, gamma,
                                                        outFeat);
}